// Actor_63333587747506
// MI455X (gfx1250) — compile-verified
//
#include <hip/hip_runtime.h>

// ---------------------------------------------------------------------------
// Problem constants (from reference): B=4096, LATENT=512, D=6, M=16, K=12,
// NB=16, R=128, HID=256, out_p = M*(1+NB+D) = 368, out_k = M*K = 192.
// ---------------------------------------------------------------------------
#define LATENTN 512
#define HIDN    256
#define KIN     518          // LATENT + D
#define BQ      16           // batch rows per block
#define MM      16           // mixtures
#define NBASIS  16
#define DD      6
#define KKN     12           // knots
#define RESN    128
#define OUTP    368          // M*(1+NB+D)
#define OUTK    192          // M*K
#define OUTT    560          // OUTP + OUTK  (= 35 tiles of 16)

typedef float v2f __attribute__((ext_vector_type(2)));
typedef float v4f __attribute__((ext_vector_type(4)));
typedef float v8f __attribute__((ext_vector_type(8)));

// fp32 WMMA 16x16x4:  D(16x16 f32, 8 VGPR) = A(16x4 f32, 2 VGPR) x B(4x16, 2 VGPR) + C
static __device__ __forceinline__ v8f wmma_f32(v2f a, v2f b, v8f c) {
    return __builtin_amdgcn_wmma_f32_16x16x4_f32(false, a, false, b, (short)0, c,
                                                 false, false);
}

// read the overlaid raw/ke buffer (rk) that lives partly in sA, partly in sExtra
static __device__ __forceinline__ float rk_read(const float* sA, const float* sExtra,
                                                int idx) {
    return (idx < BQ * HIDN) ? sA[idx] : sExtra[idx - BQ * HIDN];
}

__global__ __launch_bounds__(512) void fused_mlp_kernel(
    const float* __restrict__ latent, const float* __restrict__ intent,
    const float* __restrict__ prevv,
    const float* __restrict__ W1, const float* __restrict__ b1,
    const float* __restrict__ ln_g, const float* __restrict__ ln_b,
    const float* __restrict__ W2, const float* __restrict__ b2,
    const float* __restrict__ Wp, const float* __restrict__ bp,
    const float* __restrict__ Wk, const float* __restrict__ bk,
    const float* __restrict__ basis,
    float* __restrict__ out_logits, float* __restrict__ out_km,
    float* __restrict__ out_bw,
    float* __restrict__ ws_sig, float* __restrict__ ws_kn,
    int* __restrict__ ws_cnt)
{
    __shared__ float sBasis[NBASIS * KKN * DD];     // 1152
    __shared__ __align__(16) float sA[BQ * HIDN];   // 4096 : h / act, later rk[0:4096)
    __shared__ __align__(16) float sS[BQ * HIDN];   // 4096 : situation
    __shared__ float sExtra[BQ * OUTT - BQ * HIDN]; // 4864 : rk[4096:8960)
    __shared__ float sMu[BQ], sRstd[BQ];

    const int tid   = threadIdx.x;
    const int wave  = tid >> 5;          // 0..15
    const int lane  = tid & 31;
    const int ln16  = lane & 15;
    const int khalf = (lane < 16) ? 0 : 2;   // K sub-offset for A & B operands
    const int rhalf = (lane < 16) ? 0 : 8;   // row sub-offset for C/D
    const int r0    = blockIdx.x * BQ;

    for (int i = tid; i < NBASIS * KKN * DD; i += 512) sBasis[i] = basis[i];

    // ---------------- GEMM1: h = [latent|intent] @ W1 + b1
    {
        const int col  = wave * 16 + ln16;
        const int rowA = r0 + ln16;
        const float* __restrict__ arow = latent + (size_t)rowA * LATENTN;
        v8f acc = {};

        // main loop: K = 0..511, guard-free; A is an 8B-aligned vector load
        for (int k0 = 0; k0 < LATENTN; k0 += 4) {
            const int ka = k0 + khalf;
            const v2f a = *reinterpret_cast<const v2f*>(arow + ka);
            v2f b;
            b.x = W1[(size_t)ka * HIDN + col];
            b.y = W1[(size_t)(ka + 1) * HIDN + col];
            acc = wmma_f32(a, b, acc);
        }
        // tail step 1: K = 512..515 (intent cols 0..3), still guard-free
        {
            const int ka = LATENTN + khalf;          // 512 or 514
            v2f a, b;
            a.x = intent[rowA * DD + (ka - LATENTN)];
            a.y = intent[rowA * DD + (ka - LATENTN) + 1];
            b.x = W1[(size_t)ka * HIDN + col];
            b.y = W1[(size_t)(ka + 1) * HIDN + col];
            acc = wmma_f32(a, b, acc);
        }
        // tail step 2: K = 516..519; upper half-wave (K=518,519) is zero pad
        {
            v2f a, b;
            if (lane < 16) {
                a.x = intent[rowA * DD + 4];
                a.y = intent[rowA * DD + 5];
                b.x = W1[(size_t)516 * HIDN + col];
                b.y = W1[(size_t)517 * HIDN + col];
            } else {
                a.x = 0.f; a.y = 0.f; b.x = 0.f; b.y = 0.f;
            }
            acc = wmma_f32(a, b, acc);
        }
        for (int v = 0; v < 8; ++v)
            sA[(v + rhalf) * HIDN + col] = acc[v] + b1[col];
    }
    __syncthreads();

    // ---------------- LayerNorm stats (population variance)
    if (tid < BQ) {
        float s = 0.f, s2 = 0.f;
        for (int i = 0; i < HIDN; ++i) {
            float x = sA[tid * HIDN + i];
            s += x; s2 += x * x;
        }
        float mu  = s * (1.f / HIDN);
        float var = s2 * (1.f / HIDN) - mu * mu;
        sMu[tid]   = mu;
        sRstd[tid] = rsqrtf(var + 1e-5f);
    }
    __syncthreads();

    // ---------------- LN + ELU in place
    for (int e = 0; e < 8; ++e) {
        const int idx = tid * 8 + e;
        const int row = idx >> 8, col = idx & 255;
        float v = (sA[idx] - sMu[row]) * sRstd[row] * ln_g[col] + ln_b[col];
        sA[idx] = (v > 0.f) ? v : (expf(v) - 1.f);
    }
    __syncthreads();

    // ---------------- GEMM2: situation = act @ W2 + b2 (K = 256, A from LDS)
    {
        const int col = wave * 16 + ln16;
        v8f acc = {};
        for (int k0 = 0; k0 < HIDN; k0 += 4) {
            const int ka = k0 + khalf;
            const v2f a = *reinterpret_cast<const v2f*>(&sA[ln16 * HIDN + ka]);
            v2f b;
            b.x = W2[(size_t)ka * HIDN + col];
            b.y = W2[(size_t)(ka + 1) * HIDN + col];
            acc = wmma_f32(a, b, acc);
        }
        for (int v = 0; v < 8; ++v)
            sS[(v + rhalf) * HIDN + col] = acc[v] + b2[col];
    }
    __syncthreads();

    // ---------------- GEMM3/4: raw (368 cols) ++ ke (192 cols) = 35 tiles of 16
    for (int t = wave; t < OUTT / 16; t += 16) {
        const int c = t * 16 + ln16;
        const bool isP = (c < OUTP);     // wave-uniform: 368 = 23*16
        v8f acc = {};
        for (int k0 = 0; k0 < HIDN; k0 += 4) {
            const int ka = k0 + khalf;
            const v2f a = *reinterpret_cast<const v2f*>(&sS[ln16 * HIDN + ka]);
            v2f b;
            if (isP) {
                b.x = Wp[(size_t)ka * OUTP + c];
                b.y = Wp[(size_t)(ka + 1) * OUTP + c];
            } else {
                const int cc = c - OUTP;
                b.x = Wk[(size_t)ka * OUTK + cc];
                b.y = Wk[(size_t)(ka + 1) * OUTK + cc];
            }
            acc = wmma_f32(a, b, acc);
        }
        const float bias = isP ? bp[c] : bk[c - OUTP];
        for (int v = 0; v < 8; ++v) {
            const int idx = (v + rhalf) * OUTT + c;
            const float val = acc[v] + bias;
            if (idx < BQ * HIDN) sA[idx] = val;
            else                 sExtra[idx - BQ * HIDN] = val;
        }
    }
    __syncthreads();

    // ---------------- per-(row, mixture) post-processing
    if (tid < BQ * MM) {
        const int lr = tid >> 4, m = tid & 15;
        const int b  = r0 + lr;
        const size_t pair = (size_t)b * MM + m;
        const int base  = lr * OUTT + m * (1 + NBASIS + DD);
        const int kbase = lr * OUTT + OUTP + m * KKN;

        out_logits[pair] = rk_read(sA, sExtra, base + 0);

        float bw[NBASIS];
        for (int n = 0; n < NBASIS; ++n) {
            bw[n] = tanhf(rk_read(sA, sExtra, base + 1 + n));
            out_bw[pair * NBASIS + n] = bw[n];
        }
        for (int d = 0; d < DD; ++d)
            ws_sig[pair * DD + d] = expf(rk_read(sA, sExtra, base + 1 + NBASIS + d)) + 0.1f;

        float km[KKN];
        km[0] = 1.f; km[KKN - 1] = 1.f;
        for (int k = 1; k < KKN - 1; ++k) {
            float x = rk_read(sA, sExtra, kbase + k);
            km[k] = 1.f / (1.f + expf(-x));
        }
        for (int k = 0; k < KKN; ++k) out_km[pair * KKN + k] = km[k];

        // knots = bw @ basis  (12 x 6)
        float kn[KKN][DD];
        for (int k = 0; k < KKN; ++k)
            for (int d = 0; d < DD; ++d) {
                float acc = 0.f;
                for (int n = 0; n < NBASIS; ++n)
                    acc += bw[n] * sBasis[(n * KKN + k) * DD + d];
                kn[k][d] = acc;
            }
        // center on knot 0
        for (int d = 0; d < DD; ++d) {
            const float k0v = kn[0][d];
            for (int k = 0; k < KKN; ++k) kn[k][d] -= k0v;
        }
        // align knot 1 with previous velocity
        float fs2 = 0.f;
        for (int d = 0; d < DD; ++d) fs2 += kn[1][d] * kn[1][d];
        const float fs_mag = sqrtf(fs2);
        const float fsinv  = 1.f / (fs_mag + 1e-6f);
        float pv[DD]; float pv2 = 0.f;
        for (int d = 0; d < DD; ++d) { pv[d] = prevv[b * DD + d]; pv2 += pv[d] * pv[d]; }
        const float pvinv = 1.f / (sqrtf(pv2) + 1e-6f);
        float al[DD]; float al2 = 0.f;
        for (int d = 0; d < DD; ++d) {
            al[d] = 0.8f * pv[d] * pvinv + 0.2f * kn[1][d] * fsinv;
            al2 += al[d] * al[d];
        }
        const float alinv = 1.f / (sqrtf(al2) + 1e-6f);
        for (int d = 0; d < DD; ++d) kn[1][d] = al[d] * alinv * fs_mag;
        // blend intent into last knot
        for (int d = 0; d < DD; ++d) kn[KKN - 1][d] += 0.5f * intent[b * DD + d];

        // stable compaction: knots with mask==true first, in original order.
        // Spline indices clip to count-1, so only the first `cnt` entries matter.
        int cnt = 0;
        const size_t knb = pair * (KKN * DD);
        for (int k = 0; k < KKN; ++k) {
            if (km[k] > 0.5f) {
                for (int d = 0; d < DD; ++d)
                    ws_kn[knb + cnt * DD + d] = kn[k][d];
                ++cnt;
            }
        }
        ws_cnt[pair] = cnt;
    }
}

// ---------------------------------------------------------------------------
// Catmull-Rom sampling + sigma broadcast. One block per (b, m); 128 threads
// = 128 sample points. Results staged in LDS, then written as coalesced
// float4 (global_store_b128) bursts. This kernel writes ~402 MB and sets
// the runtime; the store path is what matters.
// ---------------------------------------------------------------------------
__global__ __launch_bounds__(128) void spline_kernel(
    const float* __restrict__ ws_kn, const int* __restrict__ ws_cnt,
    const float* __restrict__ ws_sig,
    float* __restrict__ out_mu, float* __restrict__ out_sig)
{
    __shared__ float sk[KKN * DD];
    __shared__ __align__(16) float ssig12[12];       // sigma pattern, period lcm(4,6)=12
    __shared__ __align__(16) float smu[RESN * DD];   // 768 floats = 3 KB
    __shared__ int   scnt;

    const size_t pair = blockIdx.x;
    const int t = threadIdx.x;
    if (t < KKN * DD)             sk[t]         = ws_kn[pair * (KKN * DD) + t];
    if (t >= 96 && t < 96 + 12)   ssig12[t - 96] = ws_sig[pair * DD + (t - 96) % DD];
    if (t == 127)                 scnt          = ws_cnt[pair];
    __syncthreads();

    const int cnt = scnt;
    const float x = ((float)t * (1.f / (RESN - 1))) * (float)(cnt - 1);
    int j = (int)floorf(x);
    if (j < 0) j = 0;
    if (j > cnt - 2) j = cnt - 2;
    const float u  = x - (float)j;
    const float u2 = u * u, u3 = u2 * u;
    int i0 = j - 1; if (i0 < 0) i0 = 0;
    const int i1 = j;
    int i2 = j + 1; if (i2 > cnt - 1) i2 = cnt - 1;
    int i3 = j + 2; if (i3 > cnt - 1) i3 = cnt - 1;

    for (int d = 0; d < DD; ++d) {
        const float p0 = sk[i0 * DD + d], p1 = sk[i1 * DD + d];
        const float p2 = sk[i2 * DD + d], p3 = sk[i3 * DD + d];
        smu[t * DD + d] = 0.5f * (2.f * p1 + (p2 - p0) * u +
                                  (2.f * p0 - 5.f * p1 + 4.f * p2 - p3) * u2 +
                                  (3.f * (p1 - p2) + (p3 - p0)) * u3);
    }
    __syncthreads();

    // coalesced b128 stores: 192 float4 per array per block
    const v4f* smu4  = reinterpret_cast<const v4f*>(smu);
    const v4f* ssig4 = reinterpret_cast<const v4f*>(ssig12);
    v4f* om4 = reinterpret_cast<v4f*>(out_mu)  + pair * (RESN * DD / 4);
    v4f* os4 = reinterpret_cast<v4f*>(out_sig) + pair * (RESN * DD / 4);
    for (int i = t; i < RESN * DD / 4; i += RESN) {
        om4[i] = smu4[i];
        os4[i] = ssig4[i % 3];
    }
}

extern "C" void kernel_launch(void* const* d_in, const int* in_sizes, int n_in,
                              void* d_out, int out_size, void* d_ws, size_t ws_size,
                              hipStream_t stream) {
    const float* latent = (const float*)d_in[0];
    const float* intent = (const float*)d_in[1];
    const float* prevv  = (const float*)d_in[2];
    const float* W1     = (const float*)d_in[3];
    const float* b1     = (const float*)d_in[4];
    const float* ln_g   = (const float*)d_in[5];
    const float* ln_b   = (const float*)d_in[6];
    const float* W2     = (const float*)d_in[7];
    const float* b2     = (const float*)d_in[8];
    const float* Wp     = (const float*)d_in[9];
    const float* bp     = (const float*)d_in[10];
    const float* Wk     = (const float*)d_in[11];
    const float* bk     = (const float*)d_in[12];
    const float* basis  = (const float*)d_in[13];

    const int Bn = in_sizes[0] / LATENTN;           // 4096
    const size_t nPair = (size_t)Bn * MM;

    // output layout: logits | mu_t | sigma_t | knot_mask | basis_weights
    float* out        = (float*)d_out;
    float* out_logits = out;
    float* out_mu     = out_logits + nPair;
    float* out_sig    = out_mu + nPair * RESN * DD;
    float* out_km     = out_sig + nPair * RESN * DD;
    float* out_bw     = out_km + nPair * KKN;

    // workspace: sigmas | compacted knots | counts
    float* wsf    = (float*)d_ws;
    float* ws_sig = wsf;
    float* ws_kn  = ws_sig + nPair * DD;
    int*   ws_cnt = (int*)(ws_kn + nPair * KKN * DD);

    fused_mlp_kernel<<<Bn / BQ, 512, 0, stream>>>(
        latent, intent, prevv, W1, b1, ln_g, ln_b, W2, b2, Wp, bp, Wk, bk,
        basis, out_logits, out_km, out_bw, ws_sig, ws_kn, ws_cnt);

    spline_kernel<<<(unsigned)nPair, RESN, 0, stream>>>(
        ws_kn, ws_cnt, ws_sig, out_mu, out_sig);
}